// RMSNorm_1477468749920
// MI455X (gfx1250) — compile-verified
//
#include <hip/hip_runtime.h>
#include <hip/hip_bf16.h>
#include <stdint.h>

// Fused residual-add + RMSNorm + per-group(128) dynamic FP8(E4M3) quant.
// M=16384 rows, N=4096 cols, fp32 in. Memory-bound: ~834 MB total traffic
// -> ~36us floor at 23.3 TB/s. One block (256 thr = 8 wave32) per row.
//
// CDNA5 paths: global_load_async_to_lds_b128 with PIPELINED s_wait_asynccnt
// (async loads complete in order, so chunk k only needs asynccnt <= 6-2k),
// v_cvt_pk_fp8_f32 quantization, wave32 shfl reductions, non-temporal
// stores for the 320MB of output (global L2 is 192MB).

#define N_COLS    4096
#define GROUPS    (N_COLS / 128)     // 32 quant groups per row
#define THREADS   256                // 8 wave32
#define K_ITERS   4                  // float4 chunks/thread: 4*256*4 = 4096
#define FP8_MAX_F 448.0f

typedef __attribute__((ext_vector_type(4))) float v4f;

// ---- CDNA5 async global->LDS copy (ASYNCcnt-tracked DMA-style load) ------
__device__ __forceinline__ void async_copy_b128(uint32_t lds_addr,
                                                const void* gptr) {
  // vdst = per-lane LDS byte address, vaddr = 64-bit global address
  asm volatile("global_load_async_to_lds_b128 %0, %1, off"
               :: "v"(lds_addr), "v"(gptr)
               : "memory");
}

// Partial waits: async loads retire in issue order, so waiting for
// "<= n outstanding" releases exactly the oldest (8-n) transfers.
__device__ __forceinline__ void wait_async_le(int n) {
  switch (n) {   // n is a compile-time constant after unrolling
    case 0: asm volatile("s_wait_asynccnt 0x0" ::: "memory"); break;
    case 2: asm volatile("s_wait_asynccnt 0x2" ::: "memory"); break;
    case 4: asm volatile("s_wait_asynccnt 0x4" ::: "memory"); break;
    case 6: asm volatile("s_wait_asynccnt 0x6" ::: "memory"); break;
    default: asm volatile("s_wait_asynccnt 0x0" ::: "memory"); break;
  }
}

// ---- software E4M3 fallback (only if the native builtin is unavailable) --
__device__ __forceinline__ uint32_t e4m3_sw(float f) {
  uint32_t u = __float_as_uint(f);
  uint32_t sign = (u >> 24) & 0x80u;
  uint32_t exp32 = (u >> 23) & 0xFFu;
  uint32_t man32 = u & 0x7FFFFFu;
  int e = (int)exp32 - 127 + 7;
  uint32_t mag;
  if (exp32 == 0 || e < -3) {
    mag = 0;                                   // underflow -> 0
  } else if (e <= 0) {                         // subnormal range
    uint32_t m = (man32 | 0x800000u) >> (uint32_t)(1 - e);
    mag = (m + 0x80000u) >> 20;
  } else {
    uint32_t m = man32 + 0x80000u;             // round mantissa to 3 bits
    if (m & 0x800000u) { m = 0; ++e; }
    if (e > 15) { e = 15; m = 0x600000u; }     // clamp toward 448
    mag = ((uint32_t)e << 3) | (m >> 20);
    if (mag > 0x7Eu) mag = 0x7Eu;              // no inf/nan in e4m3fn
  }
  return sign | mag;
}

__global__ __launch_bounds__(THREADS)
void fused_add_rmsnorm_fp8_quant(const float* __restrict__ x,
                                 const float* __restrict__ res,
                                 const float* __restrict__ w,
                                 uint32_t* __restrict__ q_out,    // 4 fp8/dword
                                 float*    __restrict__ scale_out,
                                 float*    __restrict__ h_out) {
  __shared__ v4f  lx[N_COLS / 4];          // 16 KB
  __shared__ v4f  lr[N_COLS / 4];          // 16 KB
  __shared__ float red[THREADS / 32];

  const int    t    = threadIdx.x;
  const int    lane = t & 31;
  const int    wave = t >> 5;
  const size_t row  = blockIdx.x;
  const size_t rowbase = row * (size_t)N_COLS;

  const v4f* xg = (const v4f*)(x + rowbase);
  const v4f* rg = (const v4f*)(res + rowbase);

  // addrspace(3) -> flat keeps the LDS byte offset in the low 32 bits
  const uint32_t lx_base = (uint32_t)(uintptr_t)&lx[0];
  const uint32_t lr_base = (uint32_t)(uintptr_t)&lr[0];

  // ---- stage the whole row (x and res) into LDS via async DMA loads ----
  // Issue order: (x0,r0),(x1,r1),(x2,r2),(x3,r3) -> 8 outstanding.
#pragma unroll
  for (int k = 0; k < K_ITERS; ++k) {
    const int idx = t + k * THREADS;            // float4 index in row
    async_copy_b128(lx_base + 16u * (uint32_t)idx, (const void*)(xg + idx));
    async_copy_b128(lr_base + 16u * (uint32_t)idx, (const void*)(rg + idx));
  }

  // ---- h = x + res, stream h out (NT), accumulate sum of squares.
  //      Pipelined: consume chunk k as soon as its pair has landed, while
  //      the remaining 6-2k transfers are still in flight. ----
  v4f   hv[K_ITERS];
  float ss = 0.0f;
#pragma unroll
  for (int k = 0; k < K_ITERS; ++k) {
    wait_async_le(6 - 2 * k);                   // oldest 2(k+1) loads done
    const int idx = t + k * THREADS;
    v4f a = lx[idx];
    v4f b = lr[idx];
    v4f h = a + b;
    hv[k] = h;
    __builtin_nontemporal_store(h, ((v4f*)(h_out + rowbase)) + idx);
    ss = fmaf(h.x, h.x, fmaf(h.y, h.y, fmaf(h.z, h.z, fmaf(h.w, h.w, ss))));
  }

  // ---- block reduction of sum-of-squares (wave32 shfl + 8-entry LDS) ----
#pragma unroll
  for (int off = 16; off >= 1; off >>= 1)
    ss += __shfl_xor(ss, off, 32);
  if (lane == 0) red[wave] = ss;
  __syncthreads();
  if (wave == 0) {
    float s = (lane < THREADS / 32) ? red[lane] : 0.0f;
#pragma unroll
    for (int off = 4; off >= 1; off >>= 1)
      s += __shfl_xor(s, off, 32);
    if (lane == 0) red[0] = s;
  }
  __syncthreads();
  const float inv_rms = rsqrtf(red[0] * (1.0f / (float)N_COLS) + 1e-6f);

  // ---- normalize, per-group(128) amax (== one wave k-slice), quantize ----
  const v4f* wg = (const v4f*)w;   // 16 KB, L2-resident across all rows
#pragma unroll
  for (int k = 0; k < K_ITERS; ++k) {
    const int idx = t + k * THREADS;
    v4f wv = wg[idx];
    v4f y  = hv[k] * inv_rms * wv;

    float am = fmaxf(fmaxf(fabsf(y.x), fabsf(y.y)),
                     fmaxf(fabsf(y.z), fabsf(y.w)));
#pragma unroll
    for (int off = 16; off >= 1; off >>= 1)
      am = fmaxf(am, __shfl_xor(am, off, 32));
    am = fmaxf(am, 1e-10f);

    const float scale = am * (1.0f / FP8_MAX_F);
    if (lane == 0)
      scale_out[row * GROUPS + (size_t)(wave + 8 * k)] = scale;

    const float inv_s = FP8_MAX_F / am;
    const float q0 = fminf(fmaxf(y.x * inv_s, -FP8_MAX_F), FP8_MAX_F);
    const float q1 = fminf(fmaxf(y.y * inv_s, -FP8_MAX_F), FP8_MAX_F);
    const float q2 = fminf(fmaxf(y.z * inv_s, -FP8_MAX_F), FP8_MAX_F);
    const float q3 = fminf(fmaxf(y.w * inv_s, -FP8_MAX_F), FP8_MAX_F);

    uint32_t packed;
#if __has_builtin(__builtin_amdgcn_cvt_pk_fp8_f32)
    int p = 0;
    p = __builtin_amdgcn_cvt_pk_fp8_f32(q0, q1, p, false);  // bytes 0..1
    p = __builtin_amdgcn_cvt_pk_fp8_f32(q2, q3, p, true);   // bytes 2..3
    packed = (uint32_t)p;
#else
    packed = e4m3_sw(q0) | (e4m3_sw(q1) << 8) |
             (e4m3_sw(q2) << 16) | (e4m3_sw(q3) << 24);
#endif
    __builtin_nontemporal_store(packed,
                                q_out + row * (size_t)(N_COLS / 4) + idx);
  }
}

extern "C" void kernel_launch(void* const* d_in, const int* in_sizes, int n_in,
                              void* d_out, int out_size, void* d_ws, size_t ws_size,
                              hipStream_t stream) {
  const float* x   = (const float*)d_in[0];
  const float* res = (const float*)d_in[1];
  const float* w   = (const float*)d_in[2];

  const int N = in_sizes[2];           // 4096 (kernel is specialized for this)
  const int M = in_sizes[0] / N;       // 16384

  // d_out = [M*N fp8 bytes][M*(N/128) f32 scales][M*N f32 residual_out]
  char*     out       = (char*)d_out;
  uint32_t* q_out     = (uint32_t*)out;
  float*    scale_out = (float*)(out + (size_t)M * (size_t)N);
  float*    h_out     = (float*)(out + (size_t)M * (size_t)N
                                     + (size_t)M * (size_t)(N / 128) * sizeof(float));

  fused_add_rmsnorm_fp8_quant<<<M, THREADS, 0, stream>>>(
      x, res, w, q_out, scale_out, h_out);
}